// GraphConv_81784767250907
// MI455X (gfx1250) — compile-verified
//
#include <hip/hip_runtime.h>
#include <hip/hip_bf16.h>

// Problem sizes (fixed by the reference)
#define N_NODES 65536
#define N_EDGES 1048576
#define D_IN    64
#define D_OUT   128

typedef __attribute__((ext_vector_type(2))) float v2f;
typedef __attribute__((ext_vector_type(8))) float v8f;

// ---------------------------------------------------------------------------
// Kernel 1: zero the agg scratch buffer (16 MB) -- must re-zero every launch.
// ---------------------------------------------------------------------------
__global__ void gc_zero_kernel(float4* __restrict__ p, int n4) {
    int i = blockIdx.x * blockDim.x + threadIdx.x;
    if (i < n4) p[i] = make_float4(0.f, 0.f, 0.f, 0.f);
}

// ---------------------------------------------------------------------------
// Kernel 2: edge aggregation. edge_dst is sorted, so each block takes a
// contiguous run of edges, accumulates per-feature in registers while dst is
// unchanged, and flushes with an atomicAdd only at segment boundaries.
// blockDim = 64 (one thread per input feature, 2 wave32s per block).
// ---------------------------------------------------------------------------
#define EDGES_PER_BLOCK 128

__global__ void gc_agg_kernel(const float* __restrict__ x,
                              const int*   __restrict__ edge_src,
                              const int*   __restrict__ edge_dst,
                              const float* __restrict__ edge_val,
                              float*       __restrict__ agg) {
    const int f  = threadIdx.x;              // feature 0..63
    const int e0 = blockIdx.x * EDGES_PER_BLOCK;
    int e_end = e0 + EDGES_PER_BLOCK;
    if (e_end > N_EDGES) e_end = N_EDGES;

    float acc = 0.f;
    int prev = -1;
    for (int e = e0; e < e_end; ++e) {
        const int d = edge_dst[e];           // uniform across block -> 1 req/wave
        if (d != prev) {
            if (prev >= 0) atomicAdd(&agg[prev * D_IN + f], acc);
            acc  = 0.f;
            prev = d;
        }
        const int   s = edge_src[e];
        const float v = edge_val[e];
        acc += x[s * D_IN + f] * v;          // 256B row: fully coalesced, L2-resident
    }
    if (prev >= 0) atomicAdd(&agg[prev * D_IN + f], acc);
}

// ---------------------------------------------------------------------------
// Kernel 3: out = agg[65536 x 64] @ W[64 x 128] in full f32 using
// V_WMMA_F32_16X16X4_F32. One wave32 computes one 16x16 output tile;
// K=64 swept in 16 steps of 4.
//
// A (16x4, M x K) per ISA: lane L<16 holds {A[L][k], A[L][k+1]},
//                          lane L>=16 holds {A[L-16][k+2], A[L-16][k+3]}.
// B (4x16, K x N) mirrored: lane L<16 holds {B[k][L], B[k+1][L]},
//                           lane L>=16 holds {B[k+2][L-16], B[k+3][L-16]}.
// C/D (16x16 f32, 8 VGPRs): VGPR r, lane L<16 -> (M=r,    N=L),
//                                    lane L>=16 -> (M=r+8, N=L-16).
// ---------------------------------------------------------------------------
__global__ void gc_gemm_kernel(const float* __restrict__ agg,
                               const float* __restrict__ W,
                               float*       __restrict__ out) {
    const int gtid  = blockIdx.x * blockDim.x + threadIdx.x;
    const int wave  = gtid >> 5;             // global wave index
    const int lane  = threadIdx.x & 31;
    const int tile_m = wave >> 3;            // 0..4095  (65536 / 16 rows)
    const int tile_n = wave & 7;             // 0..7     (128   / 16 cols)
    const int row = tile_m * 16;
    const int col = tile_n * 16;

    const int l     = lane & 15;             // position within half-wave
    const int khalf = (lane >> 4) << 1;      // 0 for lanes 0-15, 2 for 16-31

    v8f c = {};
    #pragma unroll
    for (int k = 0; k < D_IN; k += 4) {
        v2f a, b;
        const int ka = k + khalf;
        a.x = agg[(row + l) * D_IN + ka + 0];
        a.y = agg[(row + l) * D_IN + ka + 1];
        b.x = W[(ka + 0) * D_OUT + col + l];
        b.y = W[(ka + 1) * D_OUT + col + l];
        // 8 args: (neg_a, A, neg_b, B, c_mod, C, reuse_a, reuse_b)
        c = __builtin_amdgcn_wmma_f32_16x16x4_f32(
                /*neg_a=*/false, a, /*neg_b=*/false, b,
                /*c_mod=*/(short)0, c, /*reuse_a=*/false, /*reuse_b=*/false);
    }

    const int m_base = row + ((lane >> 4) << 3);   // +8 for upper half-wave
    #pragma unroll
    for (int r = 0; r < 8; ++r) {
        out[(m_base + r) * D_OUT + col + l] = c[r];
    }
}

// ---------------------------------------------------------------------------
// Launch: zero agg -> edge aggregation -> WMMA GEMM.
// ---------------------------------------------------------------------------
extern "C" void kernel_launch(void* const* d_in, const int* in_sizes, int n_in,
                              void* d_out, int out_size, void* d_ws, size_t ws_size,
                              hipStream_t stream) {
    const float* x        = (const float*)d_in[0];
    const float* weights  = (const float*)d_in[1];
    const int*   edge_src = (const int*)  d_in[2];
    const int*   edge_dst = (const int*)  d_in[3];
    const float* edge_val = (const float*)d_in[4];
    float*       out      = (float*)d_out;
    float*       agg      = (float*)d_ws;        // N_NODES * D_IN floats = 16 MB

    // 1) zero agg scratch
    {
        const int n4 = (N_NODES * D_IN) / 4;     // float4 elements
        const int threads = 256;
        const int blocks  = (n4 + threads - 1) / threads;
        gc_zero_kernel<<<blocks, threads, 0, stream>>>((float4*)agg, n4);
    }

    // 2) weighted gather + segmented scatter-add (sorted dst)
    {
        const int blocks = (N_EDGES + EDGES_PER_BLOCK - 1) / EDGES_PER_BLOCK;
        gc_agg_kernel<<<blocks, D_IN, 0, stream>>>(x, edge_src, edge_dst,
                                                   edge_val, agg);
    }

    // 3) agg @ W via f32 WMMA: 4096 x 8 tiles, 8 waves (256 thr) per block
    {
        const int total_waves = (N_NODES / 16) * (D_OUT / 16);  // 32768
        const int threads = 256;                                // 8 waves
        const int blocks  = total_waves / 8;                    // 4096
        gc_gemm_kernel<<<blocks, threads, 0, stream>>>(agg, weights, out);
    }
}